// Decoder_71116068487513
// MI455X (gfx1250) — compile-verified
//
#include <hip/hip_runtime.h>
#include <hip/hip_bf16.h>
#include <math.h>

// Problem constants (match reference)
#define B_  64
#define S_  128
#define H_  256
#define H2  512
#define H3  768
#define H4  1024
#define V_  32000
#define T_  32

typedef __attribute__((ext_vector_type(2))) float v2f;
typedef __attribute__((ext_vector_type(8))) float v8f;

// ---------------------------------------------------------------------------
// Generic fp32 WMMA GEMM:  C[M,N] = A[M,K] @ W[N,K]^T + bias
//   grid.x = N/32, grid.y = M/64, block = 256 threads (8 waves, wave32).
//   Uses V_WMMA_F32_16X16X4_F32; K must be a multiple of 64.
// A/W tiles staged through LDS (row pitch padded +2 floats: keeps v2f 8B
// alignment and breaks the 16-way bank conflict of a 64-float pitch).
// ---------------------------------------------------------------------------
#define BK 64
#define PITCH (BK + 2)   // 66 floats

__global__ __launch_bounds__(256)
void gemm64_wmma(const float* __restrict__ A, int lda,
                 const float* __restrict__ W,
                 const float* __restrict__ bias,
                 float* __restrict__ C, long ldc, int K)
{
    __shared__ float lA[64 * PITCH];
    __shared__ float lW[32 * PITCH];

    const int tid  = threadIdx.x;
    const int lane = tid & 31;
    const int wave = tid >> 5;
    const int half = lane >> 4;       // 0: lanes 0-15, 1: lanes 16-31
    const int r    = lane & 15;
    const int mt   = wave & 3;        // 4 m-tiles of 16
    const int nt   = wave >> 2;       // 2 n-tiles of 16
    const long mBlock = (long)blockIdx.y * 64;
    const long nBlock = (long)blockIdx.x * 32;

    v8f acc = {0.f, 0.f, 0.f, 0.f, 0.f, 0.f, 0.f, 0.f};

    const int ldRow = tid >> 4;          // 0..15
    const int ldCol = (tid & 15) * 4;    // 0..60

    for (int k0 = 0; k0 < K; k0 += BK) {
        // ---- stage A chunk (64 x 64), coalesced ----
        #pragma unroll
        for (int it = 0; it < 4; ++it) {
            int row = it * 16 + ldRow;
            const float* src = A + (mBlock + row) * (long)lda + k0 + ldCol;
            v2f p0 = *(const v2f*)(src);
            v2f p1 = *(const v2f*)(src + 2);
            *(v2f*)&lA[row * PITCH + ldCol]     = p0;
            *(v2f*)&lA[row * PITCH + ldCol + 2] = p1;
        }
        // ---- stage W chunk (32 x 64), coalesced ----
        #pragma unroll
        for (int it = 0; it < 2; ++it) {
            int row = it * 16 + ldRow;
            const float* src = W + (nBlock + row) * (long)K + k0 + ldCol;
            v2f p0 = *(const v2f*)(src);
            v2f p1 = *(const v2f*)(src + 2);
            *(v2f*)&lW[row * PITCH + ldCol]     = p0;
            *(v2f*)&lW[row * PITCH + ldCol + 2] = p1;
        }
        // hint the next W chunk toward the caches (global_prefetch_b8)
        if (k0 + BK < K)
            __builtin_prefetch(W + (nBlock + ldRow) * (long)K + (k0 + BK) + ldCol, 0, 1);
        __syncthreads();

        // ---- 16 WMMAs per wave (K chunk of 64, 4 per instruction) ----
        // fp32 16x16x4 fragment maps (ISA 7.12.2):
        //   A 16x4 : lanes 0-15 -> {K=0,K=1} of row M=r ; lanes 16-31 -> {K=2,K=3}
        //   B 4x16 : same striping over columns N=r
        const float* aBase = &lA[(mt * 16 + r) * PITCH + 2 * half];
        const float* wBase = &lW[(nt * 16 + r) * PITCH + 2 * half];
        #pragma unroll
        for (int kk = 0; kk < BK; kk += 4) {
            v2f af = *(const v2f*)(aBase + kk);
            v2f wf = *(const v2f*)(wBase + kk);
            acc = __builtin_amdgcn_wmma_f32_16x16x4_f32(
                false, af, false, wf, (short)0, acc, false, false);
        }
        __syncthreads();
    }

    // ---- write 16x16 tile: VGPR v -> M = 8*half + v, N = r ----
    float bv = bias ? bias[nBlock + nt * 16 + r] : 0.0f;
    #pragma unroll
    for (int v = 0; v < 8; ++v) {
        long row = mBlock + mt * 16 + half * 8 + v;
        C[row * ldc + nBlock + nt * 16 + r] = acc[v] + bv;
    }
}

// ---------------------------------------------------------------------------
// Attention: scores -> softmax -> context.  One block per batch row, 128 thr.
// score[s] = Vb + sum_j Vw[j] * tanh(qU[b,j] + encProj[b,s,j])
// ctx[b,:] written into columns [0,512) of BOTH xh buffers.
// ---------------------------------------------------------------------------
__global__ __launch_bounds__(128)
void attention_kernel(const float* __restrict__ qU,       // [B,512] (incl. Ub)
                      const float* __restrict__ encProj,  // [B,S,512] (incl. Wb)
                      const float* __restrict__ encOut,   // [B,S,512]
                      const float* __restrict__ Vw,       // [512]
                      const float* __restrict__ Vb,       // [1]
                      float* __restrict__ xhF,            // [B,1024]
                      float* __restrict__ xhB)            // [B,1024]
{
    __shared__ float qs[H2];
    __shared__ float vw[H2];
    __shared__ float wgt[S_];
    __shared__ float red[128];

    const int b   = blockIdx.x;
    const int tid = threadIdx.x;   // == s

    for (int j = tid; j < H2; j += 128) {
        qs[j] = qU[b * H2 + j];
        vw[j] = Vw[j];
    }
    __syncthreads();

    const float* ep = encProj + ((long)b * S_ + tid) * H2;
    float sc = Vb[0];
    #pragma unroll 4
    for (int j = 0; j < H2; ++j)
        sc += vw[j] * tanhf(qs[j] + ep[j]);

    // softmax over 128 scores
    red[tid] = sc;
    __syncthreads();
    for (int off = 64; off > 0; off >>= 1) {
        if (tid < off) red[tid] = fmaxf(red[tid], red[tid + off]);
        __syncthreads();
    }
    float mx = red[0];
    __syncthreads();
    float e = expf(sc - mx);
    wgt[tid] = e;
    red[tid] = e;
    __syncthreads();
    for (int off = 64; off > 0; off >>= 1) {
        if (tid < off) red[tid] += red[tid + off];
        __syncthreads();
    }
    float inv = 1.0f / red[0];
    __syncthreads();
    wgt[tid] *= inv;
    __syncthreads();

    // context: each thread owns 4 consecutive output columns (coalesced)
    const int j0 = tid * 4;
    float c0 = 0.f, c1 = 0.f, c2 = 0.f, c3 = 0.f;
    const float* eo = encOut + (long)b * S_ * H2;
    for (int s = 0; s < S_; ++s) {
        float w = wgt[s];
        const float* row = eo + s * H2 + j0;
        c0 += w * row[0]; c1 += w * row[1];
        c2 += w * row[2]; c3 += w * row[3];
    }
    float* df = xhF + b * H4 + j0;
    float* db = xhB + b * H4 + j0;
    df[0] = c0; df[1] = c1; df[2] = c2; df[3] = c3;
    db[0] = c0; db[1] = c1; db[2] = c2; db[3] = c3;
}

// ---------------------------------------------------------------------------
// Fill xh columns [512,768) = embedding(x[b]), [768,1024) = h_prev[dir]
// ---------------------------------------------------------------------------
__global__ __launch_bounds__(256)
void fill_xh(const int* __restrict__ x,
             const float* __restrict__ emb,    // [V,H]
             const float* __restrict__ hCur,   // [2,B,H]
             float* __restrict__ xhF, float* __restrict__ xhB)
{
    const int b = blockIdx.x, tid = threadIdx.x;
    const int tok = x[b];
    float ev = emb[(long)tok * H_ + tid];
    xhF[b * H4 + H2 + tid] = ev;
    xhB[b * H4 + H2 + tid] = ev;
    xhF[b * H4 + H3 + tid] = hCur[b * H_ + tid];
    xhB[b * H4 + H3 + tid] = hCur[(long)B_ * H_ + b * H_ + tid];
}

// ---------------------------------------------------------------------------
// LSTM pointwise: gates (i,f,g,o) -> h', c'; also packs out = [hf | hb]
// ---------------------------------------------------------------------------
__global__ __launch_bounds__(256)
void lstm_elem(const float* __restrict__ gatesF,  // [B,4H]
               const float* __restrict__ gatesB,
               const float* __restrict__ cCur,    // [2,B,H]
               float* __restrict__ hNext, float* __restrict__ cNext,
               float* __restrict__ outBuf)        // [B,2H]
{
    const int idx = blockIdx.x;       // 0..127
    const int d = idx >> 6, b = idx & 63;
    const int tid = threadIdx.x;      // 0..255
    const float* g = (d == 0 ? gatesF : gatesB) + b * H4;
    float gi = g[tid], gf = g[H_ + tid], gg = g[2 * H_ + tid], go = g[3 * H_ + tid];
    float cp = cCur[(long)d * B_ * H_ + b * H_ + tid];
    float si = 1.f / (1.f + expf(-gi));
    float sf = 1.f / (1.f + expf(-gf));
    float so = 1.f / (1.f + expf(-go));
    float c2 = sf * cp + si * tanhf(gg);
    float h2 = so * tanhf(c2);
    cNext[(long)d * B_ * H_ + b * H_ + tid] = c2;
    hNext[(long)d * B_ * H_ + b * H_ + tid] = h2;
    outBuf[b * H2 + d * H_ + tid] = h2;
}

// ---------------------------------------------------------------------------
// Greedy argmax over vocab row (first-max tie-break, matches jnp.argmax)
// ---------------------------------------------------------------------------
__global__ __launch_bounds__(256)
void argmax_kernel(const float* __restrict__ y, long ldy, int* __restrict__ x)
{
    __shared__ float bv[256];
    __shared__ int   bi[256];
    const int b = blockIdx.x, tid = threadIdx.x;
    const float* row = y + (long)b * ldy;
    float best = -3.402823466e38f;
    int besti = 0;
    for (int i = tid; i < V_; i += 256) {
        float v = row[i];
        if (v > best) { best = v; besti = i; }
    }
    bv[tid] = best; bi[tid] = besti;
    __syncthreads();
    for (int off = 128; off > 0; off >>= 1) {
        if (tid < off) {
            float ov = bv[tid + off]; int oi = bi[tid + off];
            if (ov > bv[tid] || (ov == bv[tid] && oi < bi[tid])) {
                bv[tid] = ov; bi[tid] = oi;
            }
        }
        __syncthreads();
    }
    if (tid == 0) x[b] = bi[0];
}

// ---------------------------------------------------------------------------
// One-time weight prep kernels
// ---------------------------------------------------------------------------
__global__ __launch_bounds__(256)
void build_ueff(const float* __restrict__ Uw, float* __restrict__ Ueff)
{
    // q = [h0,h0] => Ueff[j,k] = Uw[j,k] + Uw[j,k+256], j<512, k<256
    long i = (long)blockIdx.x * 256 + threadIdx.x;   // 512*256 elems
    int j = (int)(i >> 8), k = (int)(i & 255);
    Ueff[i] = Uw[j * H2 + k] + Uw[j * H2 + H_ + k];
}

__global__ __launch_bounds__(256)
void build_wcomb(const float* __restrict__ Wih, const float* __restrict__ Whh,
                 float* __restrict__ Wc)
{
    // Wc[j, 0:768] = Wih, Wc[j, 768:1024] = Whh  -> gates = [xin|h] @ Wc^T
    long i = (long)blockIdx.x * 256 + threadIdx.x;   // 1024*1024 elems
    int row = (int)(i >> 10), col = (int)(i & 1023);
    Wc[i] = (col < H3) ? Wih[row * H3 + col] : Whh[row * H_ + (col - H3)];
}

__global__ void init_x(int* __restrict__ x) { x[threadIdx.x] = 0; }

// ---------------------------------------------------------------------------
extern "C" void kernel_launch(void* const* d_in, const int* in_sizes, int n_in,
                              void* d_out, int out_size, void* d_ws, size_t ws_size,
                              hipStream_t stream)
{
    (void)in_sizes; (void)n_in; (void)out_size; (void)ws_size;
    const float* encOut = (const float*)d_in[0];
    const float* h0     = (const float*)d_in[1];
    const float* c0     = (const float*)d_in[2];
    const float* Uw     = (const float*)d_in[3];
    const float* Ub     = (const float*)d_in[4];
    const float* Ww     = (const float*)d_in[5];
    const float* Wb     = (const float*)d_in[6];
    const float* Vw     = (const float*)d_in[7];
    const float* Vb     = (const float*)d_in[8];
    const float* Wih_f  = (const float*)d_in[9];
    const float* Whh_f  = (const float*)d_in[10];
    const float* b_f    = (const float*)d_in[11];
    const float* Wih_b  = (const float*)d_in[12];
    const float* Whh_b  = (const float*)d_in[13];
    const float* b_b    = (const float*)d_in[14];
    const float* Fw     = (const float*)d_in[15];
    const float* Fb     = (const float*)d_in[16];
    const float* emb    = (const float*)d_in[17];
    float* out = (float*)d_out;

    // ---- workspace carve-up (floats) ----
    float* w = (float*)d_ws;
    float* encProj = w;                 w += (long)B_ * S_ * H2;   // 4,194,304
    float* Ueff    = w;                 w += (long)H2 * H_;        //   131,072
    float* WcF     = w;                 w += (long)H4 * H4;        // 1,048,576
    float* WcB     = w;                 w += (long)H4 * H4;        // 1,048,576
    float* qU      = w;                 w += (long)B_ * H2;
    float* xhF     = w;                 w += (long)B_ * H4;
    float* xhB     = w;                 w += (long)B_ * H4;
    float* gatesF  = w;                 w += (long)B_ * H4;
    float* gatesB  = w;                 w += (long)B_ * H4;
    float* outBuf  = w;                 w += (long)B_ * H2;
    float* hA      = w;                 w += 2L * B_ * H_;
    float* hB      = w;                 w += 2L * B_ * H_;
    float* cA      = w;                 w += 2L * B_ * H_;
    float* cB      = w;                 w += 2L * B_ * H_;
    int*   x       = (int*)w;

    const size_t stateBytes = 2L * B_ * H_ * sizeof(float);

    // ---- one-time prep (inside capture; deterministic) ----
    build_ueff<<<512, 256, 0, stream>>>(Uw, Ueff);
    build_wcomb<<<4096, 256, 0, stream>>>(Wih_f, Whh_f, WcF);
    build_wcomb<<<4096, 256, 0, stream>>>(Wih_b, Whh_b, WcB);
    // enc_proj = encoder_output @ Ww^T + Wb   (M=8192, N=512, K=512)
    gemm64_wmma<<<dim3(H2 / 32, (B_ * S_) / 64), 256, 0, stream>>>(
        encOut, H2, Ww, Wb, encProj, (long)H2, H2);
    init_x<<<1, 64, 0, stream>>>(x);
    hipMemcpyAsync(hA, h0, stateBytes, hipMemcpyDeviceToDevice, stream);
    hipMemcpyAsync(cA, c0, stateBytes, hipMemcpyDeviceToDevice, stream);

    float *hCur = hA, *hNxt = hB, *cCur = cA, *cNxt = cB;

    for (int t = 0; t < T_; ++t) {
        // qU = h[0] @ Ueff^T + Ub   (M=64, N=512, K=256)
        gemm64_wmma<<<dim3(H2 / 32, 1), 256, 0, stream>>>(
            hCur, H_, Ueff, Ub, qU, (long)H2, H_);
        attention_kernel<<<B_, 128, 0, stream>>>(
            qU, encProj, encOut, Vw, Vb, xhF, xhB);
        fill_xh<<<B_, 256, 0, stream>>>(x, emb, hCur, xhF, xhB);
        // gates = [ctx|emb|h] @ [Wih|Whh]^T + b   (M=64, N=1024, K=1024)
        gemm64_wmma<<<dim3(H4 / 32, 1), 256, 0, stream>>>(
            xhF, H4, WcF, b_f, gatesF, (long)H4, H4);
        gemm64_wmma<<<dim3(H4 / 32, 1), 256, 0, stream>>>(
            xhB, H4, WcB, b_b, gatesB, (long)H4, H4);
        lstm_elem<<<2 * B_, 256, 0, stream>>>(
            gatesF, gatesB, cCur, hNxt, cNxt, outBuf);
        // y[:, t, :] = out @ Fw^T + Fb   (M=64, N=32000, K=512) -> d_out
        float* yBase = out + (long)t * V_;           // row stride = T*V
        gemm64_wmma<<<dim3(V_ / 32, 1), 256, 0, stream>>>(
            outBuf, H2, Fw, Fb, yBase, (long)T_ * V_, H2);
        argmax_kernel<<<B_, 256, 0, stream>>>(yBase, (long)T_ * V_, x);

        float* th = hCur; hCur = hNxt; hNxt = th;
        float* tc = cCur; cCur = cNxt; cNxt = tc;
    }

    // final hT, cT appended after ys
    float* tail = out + (long)B_ * T_ * V_;
    hipMemcpyAsync(tail, hCur, stateBytes, hipMemcpyDeviceToDevice, stream);
    hipMemcpyAsync(tail + 2L * B_ * H_, cCur, stateBytes, hipMemcpyDeviceToDevice, stream);
}